// FlexMoERouter_26130581029444
// MI455X (gfx1250) — compile-verified
//
#include <hip/hip_runtime.h>
#include <hip/hip_bf16.h>

// Problem constants (from reference): B=4, S=2048, H=1024, E=16, TOP_K=4
#define BB   4
#define SS   2048
#define HH   1024
#define EE   16
#define KTOP 4
#define NTOK (BB * SS)      // 8192 tokens
#define NTILE (NTOK / 16)   // 512 wave-tiles (16 tokens x 16 experts each)

typedef float v2f __attribute__((ext_vector_type(2)));
typedef float v8f __attribute__((ext_vector_type(8)));

// ---------------- small helpers (16-lane group reductions, wave32) ----------------

__device__ __forceinline__ float grp_max16(float v) {
#pragma unroll
  for (int off = 8; off >= 1; off >>= 1) {
    float o = __shfl_xor(v, off, 32);
    v = fmaxf(v, o);
  }
  return v;
}

__device__ __forceinline__ float grp_sum16(float v) {
#pragma unroll
  for (int off = 8; off >= 1; off >>= 1) v += __shfl_xor(v, off, 32);
  return v;
}

// ---------------- kernels ----------------

__global__ void zero_ws_kernel(float* __restrict__ ws) {
  if (threadIdx.x < EE) ws[threadIdx.x] = 0.0f;
}

__global__ __launch_bounds__(256) void flexmoe_router_kernel(
    const float* __restrict__ x,   const float* __restrict__ img,
    const float* __restrict__ txt, const float* __restrict__ aud,
    const float* __restrict__ Wg,  const float* __restrict__ bg,
    const float* __restrict__ Wi,  const float* __restrict__ bi,
    const float* __restrict__ Wt,  const float* __restrict__ bt,
    const float* __restrict__ Wa,  const float* __restrict__ ba,
    int*   __restrict__ idx_out,    // [NTOK, 4]
    float* __restrict__ prob_out,   // [NTOK, 4]
    float* __restrict__ ws)         // [EE] per-expert prob sums
{
  const int wave = (blockIdx.x * (blockDim.x >> 5)) + (threadIdx.x >> 5);
  if (wave >= NTILE) return;  // wave-uniform; never taken with exact launch

  const int lane = threadIdx.x & 31;
  const int col  = lane & 15;   // A: row M; B/C: expert column N
  const int half = lane >> 4;   // K sub-pair select for A/B fragments
  const int tok0 = wave * 16;

  const float* ins[4] = {x,  img, txt, aud};
  const float* wts[4] = {Wg, Wi,  Wt,  Wa};

  // ---- K-swept WMMA accumulation: logits tile = sum_m  A_m(16xH) * W_m(HxE) ----
  v8f acc = {};
#pragma unroll
  for (int m = 0; m < 4; ++m) {
    // lane's A stream: row (tok0+col), starting at column 2*half
    const float* Arow = ins[m] + (size_t)(tok0 + col) * HH + 2 * half;
    const float* Wm   = wts[m];
#pragma unroll 4
    for (int k = 0; k < HH; k += 4) {
      // A fragment 16x4 f32: v0 = K (=k or k+2), v1 = K+1  -> one b64 load
      v2f a = *(const v2f*)(Arow + k);
      // B fragment 4x16 f32: v0 = row K, v1 = row K+1, striped across lanes by N
      const int kk = k + 2 * half;
      v2f b;
      b.x = Wm[kk * EE + col];
      b.y = Wm[(kk + 1) * EE + col];
      // D = A*B + C   (V_WMMA_F32_16X16X4_F32)
      acc = __builtin_amdgcn_wmma_f32_16x16x4_f32(
          /*neg_a=*/false, a, /*neg_b=*/false, b,
          /*c_mod=*/(short)0, acc, /*reuse_a=*/false, /*reuse_b=*/false);
    }
  }

  // ---- epilogue: bias, softmax over 16 experts (one 16-lane group per row) ----
  const float bias = bg[col] + bi[col] + bt[col] + ba[col];

  float esum = 0.0f;  // per-lane (expert=col) partial of sum over rows of prob
#pragma unroll
  for (int j = 0; j < 8; ++j) {
    // C layout: VGPR j, lanes 0-15 -> row j; lanes 16-31 -> row j+8
    const float logit = acc[j] + bias;
    const float mx = grp_max16(logit);
    const float p  = __expf(logit - mx);
    const float s  = grp_sum16(p);
    const float prob = p / s;
    esum += prob;

    // ---- top-4 by iterative 16-lane arg-max ----
    float v = prob;
    float tv[KTOP];
    int   ti[KTOP];
#pragma unroll
    for (int t = 0; t < KTOP; ++t) {
      float bv = v;
      int   bi2 = col;
#pragma unroll
      for (int off = 8; off >= 1; off >>= 1) {
        float ov = __shfl_xor(bv, off, 32);
        int   oi = __shfl_xor(bi2, off, 32);
        if (ov > bv || (ov == bv && oi < bi2)) { bv = ov; bi2 = oi; }
      }
      tv[t] = bv;
      ti[t] = bi2;
      if (col == bi2) v = -1.0f;  // exclude winner
    }
    const float s4 = tv[0] + tv[1] + tv[2] + tv[3];

    const int token = tok0 + j + 8 * half;
    if (col == 0) {
#pragma unroll
      for (int t = 0; t < KTOP; ++t) {
        idx_out[token * KTOP + t]  = ti[t];
        prob_out[token * KTOP + t] = tv[t] / s4;
      }
    }
  }

  // per-expert router-prob sums for aux loss (two lanes per expert contribute)
  atomicAdd(&ws[col], esum);
}

__global__ void aux_loss_kernel(const float* __restrict__ ws,
                                float* __restrict__ aux_out) {
  const int lane = threadIdx.x;
  float t = 0.0f;
  if (lane < EE) {
    const float q = ws[lane] / (float)NTOK;                 // mean over (B,S)
    t = q * __logf(q * (float)EE + 1e-9f);
  }
#pragma unroll
  for (int off = 16; off >= 1; off >>= 1) t += __shfl_xor(t, off, 32);
  if (lane == 0) *aux_out = t;
}

// ---------------- launcher ----------------

extern "C" void kernel_launch(void* const* d_in, const int* in_sizes, int n_in,
                              void* d_out, int out_size, void* d_ws, size_t ws_size,
                              hipStream_t stream) {
  const float* x   = (const float*)d_in[0];
  const float* img = (const float*)d_in[1];
  const float* txt = (const float*)d_in[2];
  const float* aud = (const float*)d_in[3];
  const float* Wg  = (const float*)d_in[4];
  const float* bg  = (const float*)d_in[5];
  const float* Wi  = (const float*)d_in[6];
  const float* bi  = (const float*)d_in[7];
  const float* Wt  = (const float*)d_in[8];
  const float* bt  = (const float*)d_in[9];
  const float* Wa  = (const float*)d_in[10];
  const float* ba  = (const float*)d_in[11];

  // Output tuple flattened: [indices (NTOK*4)] [probs (NTOK*4)] [aux (1)]
  int*   idx_out  = (int*)d_out;
  float* prob_out = (float*)d_out + (size_t)NTOK * KTOP;
  float* aux_out  = (float*)d_out + (size_t)2 * NTOK * KTOP;

  float* ws = (float*)d_ws;  // needs 16 floats

  zero_ws_kernel<<<1, 32, 0, stream>>>(ws);

  // 512 wave-tiles, 8 waves (256 threads) per block
  flexmoe_router_kernel<<<NTILE / 8, 256, 0, stream>>>(
      x, img, txt, aud, Wg, bg, Wi, bi, Wt, bt, Wa, ba,
      idx_out, prob_out, ws);

  aux_loss_kernel<<<1, 32, 0, stream>>>(ws, aux_out);
}